// GCN_30356828848616
// MI455X (gfx1250) — compile-verified
//
#include <hip/hip_runtime.h>
#include <math.h>

typedef float v2f __attribute__((ext_vector_type(2)));
typedef float v8f __attribute__((ext_vector_type(8)));

#define IN_F 512
#define HID  16

// ---------------- degree / norm ----------------
__global__ __launch_bounds__(256) void k_deg_init(float* __restrict__ deg, int N) {
    int i = blockIdx.x * 256 + threadIdx.x;
    if (i < N) deg[i] = 1.0f;              // self-loop contributes 1 to every node
}

__global__ __launch_bounds__(256) void k_deg_edges(const long long* __restrict__ ei,
                                                   float* __restrict__ deg, int E) {
    int e = blockIdx.x * 256 + threadIdx.x;
    if (e < E) {
        long long d = ei[(long long)E + e];
        atomicAdd(&deg[d], 1.0f);
    }
}

__global__ __launch_bounds__(256) void k_rsqrt(float* __restrict__ deg, int N) {
    int i = blockIdx.x * 256 + threadIdx.x;
    if (i < N) deg[i] = rsqrtf(deg[i]);    // deg >= 1 always (self loops)
}

// ---------------- h0 = x @ W1 via f32 WMMA ----------------
// One wave computes a 16(nodes) x 16(hidden) f32 tile, K=512 in steps of 4.
__global__ __launch_bounds__(256) void k_xw1_wmma(const float* __restrict__ x,
                                                  const float* __restrict__ W1,
                                                  float* __restrict__ h0, int N) {
    __shared__ float w1s[IN_F * HID];      // 32 KB: full W1
    __shared__ float xs[128 * 36];         // 18 KB: 128 rows x 32 cols, padded to 36

    const int tid = threadIdx.x;
    // stage W1 once (coalesced float4)
    {
        const float4* src = (const float4*)W1;
        float4* dst = (float4*)w1s;
#pragma unroll
        for (int i = 0; i < 8; ++i) dst[tid + 256 * i] = src[tid + 256 * i];
    }

    const int wave = tid >> 5;
    const int lane = tid & 31;
    const int m    = lane & 15;            // row (A) / col (B,C)
    const int half = lane >> 4;            // K-pair select
    const long long base_row = (long long)blockIdx.x * 128;
    const bool valid = (base_row + wave * 16) < (long long)N;

    v8f c = {};

    for (int kb0 = 0; kb0 < IN_F; kb0 += 32) {
        __syncthreads();                   // protects w1s (iter 0) and xs reuse
        // stage x[base_row..+127][kb0..kb0+31] -> xs, coalesced
#pragma unroll
        for (int i = 0; i < 4; ++i) {
            int idx = tid + 256 * i;       // 0..1023 float4 chunks
            int r   = idx >> 3;            // 0..127
            int c4  = (idx & 7) * 4;       // 0,4,..,28
            long long row = base_row + r;
            float4 v = {0.f, 0.f, 0.f, 0.f};
            if (row < (long long)N) {
                v = *(const float4*)(x + row * IN_F + kb0 + c4);
                if (kb0 + 32 < IN_F)
                    __builtin_prefetch(x + row * IN_F + kb0 + 32 + c4, 0, 1);
            }
            *(float4*)(&xs[r * 36 + c4]) = v;
        }
        __syncthreads();

        if (valid) {
            const float* xr = &xs[(wave * 16 + m) * 36];
#pragma unroll
            for (int kk = 0; kk < 32; kk += 4) {
                // A frag: lane m holds K = kk + 2*half, +1 (16x4 f32 layout)
                v2f a = *(const v2f*)(xr + kk + 2 * half);
                // B frag: lane n=m holds rows K = kb+2*half, +1 of W1
                int kw = kb0 + kk + 2 * half;
                v2f b;
                b.x = w1s[kw * HID + m];
                b.y = w1s[(kw + 1) * HID + m];
                c = __builtin_amdgcn_wmma_f32_16x16x4_f32(
                        false, a, false, b, (short)0, c, false, false);
            }
        }
    }

    if (valid) {
        // C layout: VGPR r -> M = r + 8*half, N = m
        long long orow = base_row + wave * 16 + half * 8;
#pragma unroll
        for (int r = 0; r < 8; ++r)
            h0[(orow + r) * HID + m] = c[r];
    }
}

// ---------------- layer-1 aggregation ----------------
__global__ __launch_bounds__(256) void k_self1(const float* __restrict__ h0,
                                               const float* __restrict__ dinv,
                                               float* __restrict__ h1, int N) {
    long long idx = (long long)blockIdx.x * 256 + threadIdx.x;
    if (idx < (long long)N * HID) {
        long long i = idx >> 4;
        float s = dinv[i];
        h1[idx] = h0[idx] * s * s;         // self-loop message
    }
}

__global__ __launch_bounds__(256) void k_scatter1(const long long* __restrict__ ei,
                                                  const float* __restrict__ dinv,
                                                  const float* __restrict__ h0,
                                                  float* __restrict__ h1, int E) {
    int e = blockIdx.x * 256 + threadIdx.x;
    if (e >= E) return;
    long long s = ei[e];
    long long d = ei[(long long)E + e];
    float norm = dinv[s] * dinv[d];
    const float4* hr = (const float4*)(h0 + s * HID);
    float* out = h1 + d * HID;
#pragma unroll
    for (int q = 0; q < 4; ++q) {
        float4 v = hr[q];
        atomicAdd(out + q * 4 + 0, v.x * norm);
        atomicAdd(out + q * 4 + 1, v.y * norm);
        atomicAdd(out + q * 4 + 2, v.z * norm);
        atomicAdd(out + q * 4 + 3, v.w * norm);
    }
}

// ---------------- layer-2 linear (ReLU(b1+h1) @ W2) + self term into d_out ----------------
__global__ __launch_bounds__(256) void k_lin2(const float* __restrict__ h1,
                                              const float* __restrict__ b1,
                                              const float* __restrict__ W2,
                                              const float* __restrict__ dinv,
                                              float* __restrict__ g,
                                              float* __restrict__ out, int N) {
    int i = blockIdx.x * 256 + threadIdx.x;
    if (i >= N) return;
    const float* hr = h1 + (long long)i * HID;
    float g0 = 0.f, g1 = 0.f;
#pragma unroll
    for (int f = 0; f < HID; ++f) {
        float h = hr[f] + b1[f];
        h = h > 0.f ? h : 0.f;             // ReLU (dropout is identity in eval)
        g0 += h * W2[f * 2 + 0];
        g1 += h * W2[f * 2 + 1];
    }
    g[(long long)i * 2 + 0] = g0;
    g[(long long)i * 2 + 1] = g1;
    float s = dinv[i];
    float sn = s * s;
    out[(long long)i * 2 + 0] = g0 * sn;   // self-loop message initializes accumulator
    out[(long long)i * 2 + 1] = g1 * sn;
}

__global__ __launch_bounds__(256) void k_scatter2(const long long* __restrict__ ei,
                                                  const float* __restrict__ dinv,
                                                  const float* __restrict__ g,
                                                  float* __restrict__ out, int E) {
    int e = blockIdx.x * 256 + threadIdx.x;
    if (e >= E) return;
    long long s = ei[e];
    long long d = ei[(long long)E + e];
    float norm = dinv[s] * dinv[d];
    atomicAdd(out + d * 2 + 0, g[s * 2 + 0] * norm);
    atomicAdd(out + d * 2 + 1, g[s * 2 + 1] * norm);
}

__global__ __launch_bounds__(256) void k_sigmoid(float* __restrict__ out,
                                                 const float* __restrict__ b2, int total) {
    int idx = blockIdx.x * 256 + threadIdx.x;
    if (idx < total) {
        float v = out[idx] + b2[idx & 1];
        out[idx] = 1.0f / (1.0f + __expf(-v));
    }
}

// ---------------- driver ----------------
extern "C" void kernel_launch(void* const* d_in, const int* in_sizes, int n_in,
                              void* d_out, int out_size, void* d_ws, size_t ws_size,
                              hipStream_t stream) {
    const float*      x  = (const float*)d_in[0];
    const long long*  ei = (const long long*)d_in[1];
    const float*      W1 = (const float*)d_in[2];
    const float*      b1 = (const float*)d_in[3];
    const float*      W2 = (const float*)d_in[4];
    const float*      b2 = (const float*)d_in[5];
    float* out = (float*)d_out;

    const int N = in_sizes[0] / IN_F;      // 100000
    const int E = in_sizes[1] / 2;         // 3200000

    float* ws   = (float*)d_ws;
    float* dinv = ws;                      // N
    float* h0   = dinv + N;                // 16N
    float* h1   = h0 + (size_t)16 * N;     // 16N
    float* g    = h1 + (size_t)16 * N;     // 2N

    const int TB = 256;
    int gN  = (N + TB - 1) / TB;
    int gE  = (E + TB - 1) / TB;
    int g16 = (int)(((long long)N * HID + TB - 1) / TB);
    int g2  = (2 * N + TB - 1) / TB;
    int tiles  = (N + 15) / 16;
    int gGemm  = (tiles + 7) / 8;          // 8 waves (tiles) per block

    k_deg_init <<<gN,    TB, 0, stream>>>(dinv, N);
    k_deg_edges<<<gE,    TB, 0, stream>>>(ei, dinv, E);
    k_rsqrt    <<<gN,    TB, 0, stream>>>(dinv, N);
    k_xw1_wmma <<<gGemm, TB, 0, stream>>>(x, W1, h0, N);
    k_self1    <<<g16,   TB, 0, stream>>>(h0, dinv, h1, N);
    k_scatter1 <<<gE,    TB, 0, stream>>>(ei, dinv, h0, h1, E);
    k_lin2     <<<gN,    TB, 0, stream>>>(h1, b1, W2, dinv, g, out, N);
    k_scatter2 <<<gE,    TB, 0, stream>>>(ei, dinv, g, out, E);
    k_sigmoid  <<<g2,    TB, 0, stream>>>(out, b2, 2 * N);
}